// GCN_71322226918055
// MI455X (gfx1250) — compile-verified
//
#include <hip/hip_runtime.h>
#include <hip/hip_bf16.h>

#define N_NODES 100000
#define N_EDGES 3200000
#define G_GROUPS 64

typedef __attribute__((ext_vector_type(2))) float v2f;
typedef __attribute__((ext_vector_type(8))) float v8f;

// ---------------------------------------------------------------- utilities
__global__ void gcn_zero_kernel(float* __restrict__ p, int n) {
    int i = blockIdx.x * blockDim.x + threadIdx.x;
    if (i < n) p[i] = 0.0f;
}

// ------------------------------------------------------------------ degrees
__global__ void gcn_degree_kernel(const int* __restrict__ src,
                                  const int* __restrict__ dst,
                                  float* __restrict__ out_deg,
                                  float* __restrict__ in_deg) {
    int e = blockIdx.x * blockDim.x + threadIdx.x;
    if (e < N_EDGES) {
        atomicAdd(&out_deg[src[e]], 1.0f);
        atomicAdd(&in_deg[dst[e]], 1.0f);
    }
}

// --------------------------------------------- layer-1 edge scatter (1 feat)
// h0 = in_deg ; scaled by norm_out at source, summed at dest.
__global__ void gcn_scatter1_kernel(const int* __restrict__ src,
                                    const int* __restrict__ dst,
                                    const float* __restrict__ in_deg,
                                    const float* __restrict__ out_deg,
                                    float* __restrict__ agg1) {
    int e = blockIdx.x * blockDim.x + threadIdx.x;
    if (e < N_EDGES) {
        int s = src[e];
        float v = in_deg[s] * rsqrtf(fmaxf(out_deg[s], 1.0f));
        atomicAdd(&agg1[dst[e]], v);
    }
}

// ------------------------------------- layer-1 node update (rank-1 GEMM+relu)
// h1s[n][j] = relu(agg1[n]*norm_in[n] * W1[j] + b1[j]) * norm_out[n]
// (norm_out pre-applied so the layer-2 scatter can read it directly)
__global__ void gcn_layer1_kernel(const float* __restrict__ agg1,
                                  const float* __restrict__ in_deg,
                                  const float* __restrict__ out_deg,
                                  const float* __restrict__ W1,
                                  const float* __restrict__ b1,
                                  float* __restrict__ h1s) {
    int t = blockIdx.x * blockDim.x + threadIdx.x;
    if (t >= N_NODES * 64) return;
    int n = t >> 6;
    int j = t & 63;
    float a  = agg1[n] * rsqrtf(fmaxf(in_deg[n], 1.0f));
    float h  = fmaxf(a * W1[j] + b1[j], 0.0f);
    h1s[t]   = h * rsqrtf(fmaxf(out_deg[n], 1.0f));
}

// --------------------------------------------- layer-2 edge scatter (64 feat)
// 16 lanes per edge, float4 gather + 4 f32 atomic adds each. This is the
// bandwidth-dominant kernel (~1.6 GB of traffic).
__global__ void gcn_scatter2_kernel(const int* __restrict__ src,
                                    const int* __restrict__ dst,
                                    const float* __restrict__ h1s,
                                    float* __restrict__ agg2) {
    int t = blockIdx.x * blockDim.x + threadIdx.x;          // < E*16 (51.2M)
    if (t >= N_EDGES * 16) return;
    int e = t >> 4;
    int q = (t & 15) << 2;                                   // feature*4
    int s = src[e];
    int d = dst[e];
    const float4 v = *(const float4*)(h1s + (size_t)s * 64 + q);
    float* o = agg2 + (size_t)d * 64 + q;
    atomicAdd(o + 0, v.x);
    atomicAdd(o + 1, v.y);
    atomicAdd(o + 2, v.z);
    atomicAdd(o + 3, v.w);
}

// --------------------------- layer-2 node GEMM (WMMA f32) + group-sum pooling
// h2 = (agg2 * norm_in) @ W2 + b2 ; hg[gid] += h2 (mean-divide comes later).
// Block = 128 threads = 4 waves; each wave owns a 16-wide column strip of the
// 64 output columns; block owns 16 rows (N divisible by 16 exactly).
__global__ void __launch_bounds__(128)
gcn_layer2_wmma_kernel(const float* __restrict__ agg2,
                       const float* __restrict__ in_deg,
                       const float* __restrict__ W2,
                       const float* __restrict__ b2,
                       const int* __restrict__ gid,
                       float* __restrict__ hg) {
    int wave = threadIdx.x >> 5;
    int lane = threadIdx.x & 31;
    int half = lane >> 4;          // 0: K,K+1  1: K+2,K+3
    int l    = lane & 15;
    int row_tile = blockIdx.x * 16;
    int arow = row_tile + l;       // A row (same for both lane halves)
    int col  = wave * 16 + l;      // B/D column within the 64-wide output
    float nin = rsqrtf(fmaxf(in_deg[arow], 1.0f));

    v8f acc = {};
    const float* arow_p = agg2 + (size_t)arow * 64;
#pragma unroll
    for (int k0 = 0; k0 < 64; k0 += 4) {
        int kk = k0 + half * 2;
        v2f a, b;
        a.x = arow_p[kk]     * nin;
        a.y = arow_p[kk + 1] * nin;
        b.x = W2[(size_t)kk       * 64 + col];
        b.y = W2[(size_t)(kk + 1) * 64 + col];
        acc = __builtin_amdgcn_wmma_f32_16x16x4_f32(
            false, a, false, b, (short)0, acc, false, false);
    }

    float bias = b2[col];
#pragma unroll
    for (int r = 0; r < 8; ++r) {
        int orow = row_tile + half * 8 + r;   // D layout: VGPR r = row r / 8+r
        float v = acc[r] + bias;
        atomicAdd(&hg[(size_t)gid[orow] * 64 + col], v);
    }
}

// ------------------------------------------------------------- group counts
__global__ void gcn_counts_kernel(const int* __restrict__ gid,
                                  float* __restrict__ counts) {
    int n = blockIdx.x * blockDim.x + threadIdx.x;
    if (n < N_NODES) atomicAdd(&counts[gid[n]], 1.0f);
}

__global__ void gcn_pooldiv_kernel(float* __restrict__ hg,
                                   const float* __restrict__ counts) {
    int t = blockIdx.x * blockDim.x + threadIdx.x;   // G*64
    if (t < G_GROUPS * 64) {
        hg[t] = hg[t] / fmaxf(counts[t >> 6], 1.0f);
    }
}

// ----------------------------------------------- MLP dense layer (WMMA f32)
// Y[M=64, Ncols] = act(X[64,K] @ W[K,Ncols] + bias). One wave per 16x16 tile.
__global__ void __launch_bounds__(32)
gcn_mlp_wmma_kernel(const float* __restrict__ X,
                    const float* __restrict__ W,
                    const float* __restrict__ bias,
                    float* __restrict__ Y,
                    int K, int Ncols, int relu) {
    int nt = blockIdx.x;          // column tile
    int mt = blockIdx.y;          // row tile
    int lane = threadIdx.x;
    int half = lane >> 4;
    int l    = lane & 15;
    int row = mt * 16 + l;
    int col = nt * 16 + l;

    v8f acc = {};
    const float* xrow = X + (size_t)row * K;
    for (int k0 = 0; k0 < K; k0 += 4) {
        int kk = k0 + half * 2;
        v2f a, b;
        a.x = xrow[kk];
        a.y = xrow[kk + 1];
        b.x = W[(size_t)kk       * Ncols + col];
        b.y = W[(size_t)(kk + 1) * Ncols + col];
        acc = __builtin_amdgcn_wmma_f32_16x16x4_f32(
            false, a, false, b, (short)0, acc, false, false);
    }

    float bv = bias[col];
#pragma unroll
    for (int r = 0; r < 8; ++r) {
        int orow = mt * 16 + half * 8 + r;
        float v = acc[r] + bv;
        if (relu) v = fmaxf(v, 0.0f);
        Y[(size_t)orow * Ncols + col] = v;
    }
}

// ------------------------------------- final ragged layer (512->10) + softmax
__global__ void gcn_final_kernel(const float* __restrict__ X,
                                 const float* __restrict__ W5,
                                 const float* __restrict__ b5,
                                 float* __restrict__ out) {
    int row = blockIdx.x * blockDim.x + threadIdx.x;
    if (row >= G_GROUPS) return;
    float acc[10];
#pragma unroll
    for (int j = 0; j < 10; ++j) acc[j] = b5[j];
    const float* xr = X + (size_t)row * 512;
    for (int k = 0; k < 512; ++k) {
        float x = xr[k];
#pragma unroll
        for (int j = 0; j < 10; ++j) acc[j] += x * W5[k * 10 + j];
    }
    float m = acc[0];
#pragma unroll
    for (int j = 1; j < 10; ++j) m = fmaxf(m, acc[j]);
    float s = 0.0f;
#pragma unroll
    for (int j = 0; j < 10; ++j) { acc[j] = expf(acc[j] - m); s += acc[j]; }
    float inv = 1.0f / s;
#pragma unroll
    for (int j = 0; j < 10; ++j) out[row * 10 + j] = acc[j] * inv;
}

// =========================================================================
extern "C" void kernel_launch(void* const* d_in, const int* in_sizes, int n_in,
                              void* d_out, int out_size, void* d_ws, size_t ws_size,
                              hipStream_t stream) {
    const int*   edge_src = (const int*)  d_in[0];
    const int*   edge_dst = (const int*)  d_in[1];
    const int*   node_gid = (const int*)  d_in[2];
    const float* W1  = (const float*)d_in[3];
    const float* b1  = (const float*)d_in[4];
    const float* W2  = (const float*)d_in[5];
    const float* b2  = (const float*)d_in[6];
    const float* Wc1 = (const float*)d_in[7];
    const float* bc1 = (const float*)d_in[8];
    const float* Wc2 = (const float*)d_in[9];
    const float* bc2 = (const float*)d_in[10];
    const float* Wc3 = (const float*)d_in[11];
    const float* bc3 = (const float*)d_in[12];
    const float* Wc4 = (const float*)d_in[13];
    const float* bc4 = (const float*)d_in[14];
    const float* Wc5 = (const float*)d_in[15];
    const float* bc5 = (const float*)d_in[16];
    float* out = (float*)d_out;

    // ---- workspace layout (floats, all offsets 16B-aligned) ----
    float* ws      = (float*)d_ws;
    float* in_deg  = ws;
    float* out_deg = in_deg  + N_NODES;
    float* agg1    = out_deg + N_NODES;
    float* h1s     = agg1    + N_NODES;            // N*64
    float* agg2    = h1s     + (size_t)N_NODES * 64;
    float* hg      = agg2    + (size_t)N_NODES * 64;   // G*64
    float* counts  = hg      + G_GROUPS * 64;          // G
    float* x1      = counts  + G_GROUPS;               // 64*512
    float* x2      = x1      + 64 * 512;               // 64*1024
    float* x3      = x2      + 64 * 1024;              // 64*1024
    float* x4      = x3      + 64 * 1024;              // 64*512

    const int TB = 256;
    auto blocks = [](long long n, int tb) { return (unsigned)((n + tb - 1) / tb); };

    // zero accumulators: [in_deg,out_deg,agg1], agg2, [hg,counts]
    gcn_zero_kernel<<<blocks(3LL * N_NODES, TB), TB, 0, stream>>>(in_deg, 3 * N_NODES);
    gcn_zero_kernel<<<blocks((long long)N_NODES * 64, TB), TB, 0, stream>>>(agg2, N_NODES * 64);
    gcn_zero_kernel<<<blocks(G_GROUPS * 64 + G_GROUPS, TB), TB, 0, stream>>>(hg, G_GROUPS * 64 + G_GROUPS);

    // degrees
    gcn_degree_kernel<<<blocks(N_EDGES, TB), TB, 0, stream>>>(edge_src, edge_dst, out_deg, in_deg);

    // layer 1: edge scatter (1 feature) then node rank-1 update
    gcn_scatter1_kernel<<<blocks(N_EDGES, TB), TB, 0, stream>>>(edge_src, edge_dst, in_deg, out_deg, agg1);
    gcn_layer1_kernel<<<blocks((long long)N_NODES * 64, TB), TB, 0, stream>>>(
        agg1, in_deg, out_deg, W1, b1, h1s);

    // layer 2: edge scatter (64 features, bandwidth-dominant)
    gcn_scatter2_kernel<<<blocks((long long)N_EDGES * 16, TB), TB, 0, stream>>>(
        edge_src, edge_dst, h1s, agg2);

    // layer 2 node GEMM via v_wmma_f32_16x16x4_f32 + fused group pooling
    gcn_layer2_wmma_kernel<<<N_NODES / 16, 128, 0, stream>>>(
        agg2, in_deg, W2, b2, node_gid, hg);

    // group mean
    gcn_counts_kernel<<<blocks(N_NODES, TB), TB, 0, stream>>>(node_gid, counts);
    gcn_pooldiv_kernel<<<blocks(G_GROUPS * 64, TB), TB, 0, stream>>>(hg, counts);

    // MLP head, each layer as 16x16 WMMA tiles (M=64 -> 4 row tiles)
    gcn_mlp_wmma_kernel<<<dim3(512 / 16, 4), 32, 0, stream>>>(hg, Wc1, bc1, x1, 64,   512, 1);
    gcn_mlp_wmma_kernel<<<dim3(1024 / 16, 4), 32, 0, stream>>>(x1, Wc2, bc2, x2, 512, 1024, 1);
    gcn_mlp_wmma_kernel<<<dim3(1024 / 16, 4), 32, 0, stream>>>(x2, Wc3, bc3, x3, 1024, 1024, 1);
    gcn_mlp_wmma_kernel<<<dim3(512 / 16, 4), 32, 0, stream>>>(x3, Wc4, bc4, x4, 1024, 512, 1);

    // final 512->10 + softmax
    gcn_final_kernel<<<2, 32, 0, stream>>>(x4, Wc5, bc5, out);
}